// ManualGCNLayer_16071767622244
// MI455X (gfx1250) — compile-verified
//
#include <hip/hip_runtime.h>
#include <hip/hip_bf16.h>

#define N_NODES 100000
#define N_EDGES 1600000
#define IN_DIM  64
#define OUT_DIM 64

typedef __attribute__((ext_vector_type(2))) float v2f;
typedef __attribute__((ext_vector_type(8))) float v8f;

// ---------------------------------------------------------------------------
// Zero the workspace (agg + deg). Graph-capture safe (plain kernel).
// ---------------------------------------------------------------------------
__global__ void gcn_zero_ws(float* __restrict__ p, int n) {
    int i = blockIdx.x * blockDim.x + threadIdx.x;
    if (i < n) p[i] = 0.0f;
}

// ---------------------------------------------------------------------------
// Edge scatter: one wave32 per edge. Lane L handles features {2L, 2L+1}
// via a coalesced float2 gather from x[col] and two f32 atomics into
// agg[row]. Lane 0 bumps the (float) degree counter.
// ---------------------------------------------------------------------------
__global__ __launch_bounds__(256) void gcn_scatter(const float* __restrict__ x,
                                                   const long long* __restrict__ ei,
                                                   float* __restrict__ agg,
                                                   float* __restrict__ deg) {
    int gtid = blockIdx.x * blockDim.x + threadIdx.x;
    int edge = gtid >> 5;          // global wave id == edge id
    int lane = threadIdx.x & 31;
    if (edge >= N_EDGES) return;

    long long row = ei[edge];               // destination node
    long long col = ei[N_EDGES + edge];     // source node

    const float2* src = (const float2*)(x + (size_t)col * IN_DIM);
    float2 v = src[lane];

    float* dst = agg + (size_t)row * IN_DIM + lane * 2;
    atomicAdd(dst,     v.x);
    atomicAdd(dst + 1, v.y);
    if (lane == 0) atomicAdd(deg + row, 1.0f);
}

// ---------------------------------------------------------------------------
// Fused normalize + linear: out[n,:] = (agg[n,:]/max(deg[n],1)) @ W^T + b
// Block = 128 threads (4 waves). Each wave computes a 16x16 tile of the
// output with V_WMMA_F32_16X16X4_F32 over K=64 (16 WMMA ops per wave).
// W (64x64 f32, 16KB) and bias are staged in LDS once per block.
//
// WMMA 32-bit operand layout (wave32):
//   A (16x4):  lane l<16 holds A[M=l][K=k..k+1], lane l+16 holds K=k+2..k+3
//   B (4x16):  lane l<16 holds B[K=k..k+1][N=l], lane l+16 holds K=k+2..k+3
//   D (16x16): VGPR r: lanes 0-15 -> D[M=r][N=lane], lanes 16-31 -> D[M=r+8]
// Since out = A_norm * W^T, B[k][n] = W[n][k]: a float2 load from row n of W
// at offset k (+2 for the upper half-wave) is exactly the B fragment.
// N_NODES = 100000 = 16 * 6250, so there is no partial tile.
// ---------------------------------------------------------------------------
__global__ __launch_bounds__(128) void gcn_gemm(const float* __restrict__ agg,
                                                const float* __restrict__ deg,
                                                const float* __restrict__ W,
                                                const float* __restrict__ bias,
                                                float* __restrict__ out) {
    __shared__ float Wlds[OUT_DIM * IN_DIM];
    __shared__ float blds[OUT_DIM];

    int tid = threadIdx.x;

    // cooperative 128-bit staging of W into LDS
    const float4* W4  = (const float4*)W;
    float4*       Wl4 = (float4*)Wlds;
    for (int i = tid; i < (OUT_DIM * IN_DIM) / 4; i += 128) Wl4[i] = W4[i];
    if (tid < OUT_DIM) blds[tid] = bias[tid];
    __syncthreads();

    const int wave = tid >> 5;          // 0..3 -> output-column tile
    const int lane = tid & 31;
    const int half = lane >> 4;         // 0: K lo pair, 1: K hi pair
    const int l    = lane & 15;

    const int node0 = blockIdx.x * 16;
    const int node  = node0 + l;        // A-matrix row for this lane

    // fold degree normalization into the A fragment
    const float rd = 1.0f / fmaxf(deg[node], 1.0f);

    const float* arow = agg  + (size_t)node * IN_DIM + half * 2;
    const float* wrow = Wlds + (wave * 16 + l) * IN_DIM + half * 2;

    v8f c = {0.f, 0.f, 0.f, 0.f, 0.f, 0.f, 0.f, 0.f};

#pragma unroll
    for (int k = 0; k < IN_DIM; k += 4) {
        float2 av = *(const float2*)(arow + k);
        float2 bv = *(const float2*)(wrow + k);
        v2f a, b;
        a[0] = av.x * rd; a[1] = av.y * rd;
        b[0] = bv.x;      b[1] = bv.y;
        // (neg_a, A, neg_b, B, c_mod, C, reuse_a, reuse_b)
        c = __builtin_amdgcn_wmma_f32_16x16x4_f32(false, a, false, b,
                                                  (short)0, c, false, false);
    }

    const int ocol = wave * 16 + l;
    const float bv = blds[ocol];
#pragma unroll
    for (int r = 0; r < 8; ++r) {
        const int m = r + half * 8;     // row within tile
        out[(size_t)(node0 + m) * OUT_DIM + ocol] = c[r] + bv;
    }
}

// ---------------------------------------------------------------------------
// Host-side launcher (graph-capture safe: kernels only, all on `stream`)
// inputs: d_in[0]=x f32[100000*64], d_in[1]=edge_index i64[2*1600000],
//         d_in[2]=W f32[64*64], d_in[3]=b f32[64]
// d_ws layout: agg f32[100000*64] | deg f32[100000]   (26.0 MB)
// ---------------------------------------------------------------------------
extern "C" void kernel_launch(void* const* d_in, const int* in_sizes, int n_in,
                              void* d_out, int out_size, void* d_ws, size_t ws_size,
                              hipStream_t stream) {
    const float*     x  = (const float*)d_in[0];
    const long long* ei = (const long long*)d_in[1];
    const float*     W  = (const float*)d_in[2];
    const float*     b  = (const float*)d_in[3];
    float*           out = (float*)d_out;

    float* agg = (float*)d_ws;
    float* deg = agg + (size_t)N_NODES * IN_DIM;

    // 1) zero agg + deg
    const int zn = N_NODES * (IN_DIM + 1);
    gcn_zero_ws<<<(zn + 255) / 256, 256, 0, stream>>>(agg, zn);

    // 2) scatter-add along edges (one wave32 per edge)
    const long long sthreads = (long long)N_EDGES * 32;
    gcn_scatter<<<(int)((sthreads + 255) / 256), 256, 0, stream>>>(x, ei, agg, deg);

    // 3) fused normalize + linear with WMMA
    gcn_gemm<<<N_NODES / 16, 128, 0, stream>>>(agg, deg, W, b, out);
}